// Network_74457553043499
// MI455X (gfx1250) — compile-verified
//
#include <hip/hip_runtime.h>
#include <stdint.h>

// FDTD fused single-step: E (Ampere, bc+frog) then Hx/Hy (Faraday, bc+frog)
// 4096x4096 f32, memory-bound -> single fused pass.
// Staging: TDM tensor_load_to_lds (one DMA per field) on interior tiles,
// per-lane clamped global_load_async_to_lds on edge tiles.
// Outputs: non-temporal stores (keep L2 for halo-reused inputs).

#define N      4096
#define TILE   32
#define RIN    40           // input tile rows/cols (TILE + 2*4 halo)
#define PIN    40           // packed pitch (TDM writes rows contiguously)
#define RE     36           // E tile rows/cols (TILE + 2*2 halo)
#define PE     37           // LDS pitch for E tile
#define SZIN   (RIN * PIN)  // 1600 floats
#define SZE    (RE * PE)    // 1332 floats
#define SMEM_FLOATS (3 * SZIN + SZE)
#define NTHREADS 256

typedef unsigned int u32;
typedef __attribute__((ext_vector_type(4))) u32 u32x4;
typedef __attribute__((ext_vector_type(8))) u32 u32x8;

__device__ __forceinline__ uint32_t lds_lo32(const void* p) {
  // low 32 bits of a generic pointer to LDS == workgroup-relative LDS byte address
  return (uint32_t)(uintptr_t)p;
}

// One TDM descriptor: 2D tile RINxRIN of f32, row stride N, global -> LDS.
// D# group0: count=1 | lds_addr | global_addr[56:0] | type=2
// D# group1: data_size=4B | tensor_dim0/1 | tile_dim0/1 | tensor_dim0_stride
__device__ __forceinline__ void tdm_load_tile(const float* tile_origin,
                                              uint32_t lds_byte_addr) {
  const uint64_t ga = (uint64_t)(uintptr_t)tile_origin;
  const u32x4 g0 = {
      0x1u,                                                  // count=1, user D#
      lds_byte_addr,                                         // lds_addr
      (u32)ga,                                               // global_addr[31:0]
      (u32)((ga >> 32) & 0x1FFFFFFu) | (2u << 30)            // [56:32] | type=2
  };
  const u32x8 g1 = {
      2u << 16,                                              // data_size = 4 bytes
      ((u32)N & 0xFFFFu) << 16,                              // tensor_dim0[15:0]
      ((u32)N >> 16) | (((u32)N & 0xFFFFu) << 16),           // dim0 hi | dim1 lo
      ((u32)N >> 16) | ((u32)RIN << 16),                     // dim1 hi | tile_dim0
      (u32)RIN,                                              // tile_dim1 | tile_dim2=0
      (u32)N,                                                // tensor_dim0_stride lo
      0u,                                                    // stride hi | dim1_stride lo
      0u                                                     // dim1_stride hi
  };
  asm volatile("tensor_load_to_lds %0, %1" :: "s"(g0), "s"(g1) : "memory");
}

__global__ __launch_bounds__(NTHREADS) void fdtd_fused_kernel(
    const float* __restrict__ X1, const float* __restrict__ X2,
    const float* __restrict__ X3, const float* __restrict__ W1,
    float* __restrict__ outE, float* __restrict__ outHx,
    float* __restrict__ outHy)
{
  extern __shared__ float smem[];
  float* sX1 = smem;
  float* sX2 = smem + SZIN;
  float* sX3 = smem + 2 * SZIN;
  float* sE  = smem + 3 * SZIN;

  const int tid = threadIdx.x;
  const int gi0 = blockIdx.y * TILE;   // row (i) origin of output tile
  const int gj0 = blockIdx.x * TILE;   // col (j) origin of output tile

  const uint32_t smemA = lds_lo32(smem);

  // ---- Stage 1: stage three (TILE+8)^2 input tiles into LDS ----
  const bool interior = (gi0 >= 4) && (gi0 + TILE + 4 <= N) &&
                        (gj0 >= 4) && (gj0 + TILE + 4 <= N);
  if (interior) {
    // Whole halo window in-range: one TDM DMA per field, issued by wave 0 only
    // (waves 1..7 skip via execz branch; TDM itself ignores EXEC).
    if (tid < 32) {
      const size_t org = (size_t)(gi0 - 4) * N + (gj0 - 4);
      tdm_load_tile(X1 + org, smemA + 0u * SZIN * 4u);
      tdm_load_tile(X2 + org, smemA + 1u * SZIN * 4u);
      tdm_load_tile(X3 + org, smemA + 2u * SZIN * 4u);
      __builtin_amdgcn_s_wait_tensorcnt(0);
    }
  } else {
    // Edge tiles: per-lane async global->LDS with clamped indices; clamped
    // values are never consumed by a valid piecewise formula.
    for (int f = 0; f < 3; ++f) {
      const float* src = (f == 0) ? X1 : (f == 1) ? X2 : X3;
      const uint32_t ldsBase = smemA + (uint32_t)(f * SZIN) * 4u;
      for (int idx = tid; idx < RIN * RIN; idx += NTHREADS) {
        int r = idx / RIN, c = idx - r * RIN;
        int gi = gi0 - 4 + r; gi = gi < 0 ? 0 : (gi > N - 1 ? N - 1 : gi);
        int gj = gj0 - 4 + c; gj = gj < 0 ? 0 : (gj > N - 1 ? N - 1 : gj);
        uint32_t voff  = (uint32_t)((gi << 12) + gj) * 4u;      // byte offset
        uint32_t laddr = ldsBase + (uint32_t)idx * 4u;          // packed pitch
        asm volatile("global_load_async_to_lds_b32 %0, %1, %2"
                     :: "v"(laddr), "v"(voff), "s"(src) : "memory");
      }
    }
    asm volatile("s_wait_asynccnt 0" ::: "memory");
  }
  __syncthreads();

  // ---- Coefficients ----
  const float w  = W1[0];
  const float f0 = (w - 1.0f) / 3.0f;
  const float f1 = -w;
  const float f2 = w;
  const float f3 = (1.0f - w) / 3.0f;
  const float C  = (float)((double)(N - 1) / 400.0);   // DT/DX = DT/DY, Z=1

  // ---- Stage 2: compute E tile with +/-2 halo into LDS ----
  // E regions: outer ring -> X1 ; ring i,j in {1,N-2} -> E1 (2-tap) ;
  // interior [2,N-3]^2 -> E1 + 4-tap f4 correction.
  for (int idx = tid; idx < RE * RE; idx += NTHREADS) {
    int er = idx / RE, ec = idx - er * RE;
    int i = gi0 - 2 + er, j = gj0 - 2 + ec;     // global coords (may be halo)
    int ir = er + 2, ic = ec + 2;               // input-LDS coords
    float e;
    if (i <= 0 || i >= N - 1 || j <= 0 || j >= N - 1) {
      e = sX1[ir * PIN + ic];                   // boundary (and unused out-of-grid halo)
    } else {
      e = sX1[ir * PIN + ic]
        + C * (sX3[ir * PIN + ic] - sX3[(ir - 1) * PIN + ic])
        - C * (sX2[ir * PIN + ic] - sX2[ir * PIN + ic - 1]);
      if (i >= 2 && i <= N - 3 && j >= 2 && j <= N - 3) {
        e += C * (f0 * sX3[(ir - 2) * PIN + ic] + f1 * sX3[(ir - 1) * PIN + ic]
                + f2 * sX3[ir * PIN + ic]       + f3 * sX3[(ir + 1) * PIN + ic])
           - C * (f0 * sX2[ir * PIN + ic - 2]   + f1 * sX2[ir * PIN + ic - 1]
                + f2 * sX2[ir * PIN + ic]       + f3 * sX2[ir * PIN + ic + 1]);
      }
    }
    sE[er * PE + ec] = e;
  }
  __syncthreads();

  // ---- Stage 3: outputs (E, Hx, Hy) with non-temporal stores ----
  for (int idx = tid; idx < TILE * TILE; idx += NTHREADS) {
    int oi = idx / TILE, oj = idx - oi * TILE;
    int i = gi0 + oi, j = gj0 + oj;
    int ir = oi + 4, ic = oj + 4;   // input-LDS coords
    int er = oi + 2, ec = oj + 2;   // E-LDS coords
    size_t g = (size_t)i * N + j;

    // E
    __builtin_nontemporal_store(sE[er * PE + ec], &outE[g]);

    // Hx: Hx1 on [1,N-2]x[0,N-2]; 4-tap E correction on [2,N-3]x[1,N-3]
    float x2 = sX2[ir * PIN + ic];
    float hx;
    if (i >= 1 && i <= N - 2 && j <= N - 2) {
      float hx1 = x2 - C * (sX1[ir * PIN + ic + 1] - sX1[ir * PIN + ic]);
      if (i >= 2 && i <= N - 3 && j >= 1 && j <= N - 3) {
        hx1 -= C * (f0 * sE[er * PE + ec - 1] + f1 * sE[er * PE + ec]
                  + f2 * sE[er * PE + ec + 1] + f3 * sE[er * PE + ec + 2]);
      }
      hx = hx1;
    } else {
      hx = x2;
    }
    __builtin_nontemporal_store(hx, &outHx[g]);

    // Hy: Hy1 on [0,N-2]x[1,N-2]; 4-tap E correction on [1,N-3]x[2,N-3]
    float x3 = sX3[ir * PIN + ic];
    float hy;
    if (i <= N - 2 && j >= 1 && j <= N - 2) {
      float hy1 = x3 + C * (sX1[(ir + 1) * PIN + ic] - sX1[ir * PIN + ic]);
      if (i >= 1 && i <= N - 3 && j >= 2 && j <= N - 3) {
        hy1 += C * (f0 * sE[(er - 1) * PE + ec] + f1 * sE[er * PE + ec]
                  + f2 * sE[(er + 1) * PE + ec] + f3 * sE[(er + 2) * PE + ec]);
      }
      hy = hy1;
    } else {
      hy = x3;
    }
    __builtin_nontemporal_store(hy, &outHy[g]);
  }
}

extern "C" void kernel_launch(void* const* d_in, const int* in_sizes, int n_in,
                              void* d_out, int out_size, void* d_ws, size_t ws_size,
                              hipStream_t stream) {
  const float* X1 = (const float*)d_in[0];
  const float* X2 = (const float*)d_in[1];
  const float* X3 = (const float*)d_in[2];
  const float* W1 = (const float*)d_in[3];

  float* outE  = (float*)d_out;
  float* outHx = outE + (size_t)N * N;
  float* outHy = outHx + (size_t)N * N;

  dim3 grid(N / TILE, N / TILE);
  size_t shmem = SMEM_FLOATS * sizeof(float);
  fdtd_fused_kernel<<<grid, NTHREADS, shmem, stream>>>(X1, X2, X3, W1,
                                                       outE, outHx, outHy);
}